// VectorQuantizer_15771119911137
// MI455X (gfx1250) — compile-verified
//
#include <hip/hip_runtime.h>

typedef __attribute__((ext_vector_type(2))) float v2f;
typedef __attribute__((ext_vector_type(8))) float v8f;

#define VQ_D        64                      // embedding dim
#define VQ_K        1024                    // codebook size
#define VQ_NT       (VQ_K / 16)             // 64 column tiles
#define VQ_KSTEPS   17                      // ceil(68/4): 16 data k-steps + 1 (||e||^2 row + pad)
#define FRAG_FLOATS (VQ_KSTEPS * 32 * 2)    // 1088 floats per column-tile fragment set
#define BP_FLOATS   (VQ_NT * FRAG_FLOATS)   // 69632 floats total (278528 B)
#define CHUNK_T     8                       // column tiles staged in LDS at a time
#define CHUNK_FLOATS (CHUNK_T * FRAG_FLOATS) // 8704 floats = 34816 B LDS

// ---------------------------------------------------------------------------
// Prep: build augmented B operand in WMMA-fragment layout.
//   layout: bp[t][s][lane][j], fragment element (lane, j) of k-step s, tile t:
//     k = 4*s + 2*(lane>>4) + j ; n = 16*t + (lane&15)
//     value = (k < 64) ? -2*emb[n][k] : (k == 64) ? ||e_n||^2 : 0
// ---------------------------------------------------------------------------
__global__ void vq_prep_b(const float* __restrict__ emb, float* __restrict__ bp) {
    int tid = blockIdx.x * blockDim.x + threadIdx.x;
    if (tid >= BP_FLOATS) return;
    int t    = tid / FRAG_FLOATS;
    int r    = tid % FRAG_FLOATS;
    int s    = r / 64;
    int r2   = r % 64;
    int lane = r2 >> 1;
    int j    = r2 & 1;
    int k    = 4 * s + 2 * (lane >> 4) + j;
    int n    = 16 * t + (lane & 15);
    float v  = 0.0f;
    if (k < VQ_D) {
        v = -2.0f * emb[n * VQ_D + k];
    } else if (k == VQ_D) {
        float acc = 0.0f;
        for (int q = 0; q < VQ_D; ++q) { float e = emb[n * VQ_D + q]; acc += e * e; }
        v = acc;
    }
    bp[tid] = v;
}

__global__ void vq_zero_counts(unsigned int* __restrict__ counts) {
    int tid = blockIdx.x * blockDim.x + threadIdx.x;
    if (tid < VQ_K) counts[tid] = 0u;
}

// ---------------------------------------------------------------------------
// Main: 8 waves / block, one 16-row tile per wave. Distances come straight out
// of the WMMA accumulator (dist = ||e||^2 - 2 x.e); per-lane running argmin,
// then 16-lane shfl_xor argmin reduce, then gather winning embeddings.
// ---------------------------------------------------------------------------
__global__ __launch_bounds__(256)
void vq_main(const float* __restrict__ x, const float* __restrict__ emb,
             const float* __restrict__ bp, float* __restrict__ out,
             unsigned int* __restrict__ counts) {
    __shared__ float lds[CHUNK_FLOATS];

    const int tid  = threadIdx.x;
    const int lane = tid & 31;
    const int wave = tid >> 5;
    const int hi   = lane >> 4;     // which half of the wave
    const int ln   = lane & 15;
    const int m0   = (blockIdx.x * 8 + wave) * 16;
    const int row  = m0 + ln;

    // Preload A fragments: 16 k-steps covering x[row][0..63].
    // A layout (f32 16x4): lanes 0-15 hold K = 4s+{0,1}; lanes 16-31 K = 4s+{2,3}.
    v2f a[16];
    const float* xrow = x + (size_t)row * VQ_D;
    #pragma unroll
    for (int s = 0; s < 16; ++s)
        a[s] = *(const v2f*)(xrow + 4 * s + 2 * hi);
    // Constant final fragment: column of ones at k==64 (picks up ||e||^2 row of B).
    v2f afin; afin[0] = (hi == 0) ? 1.0f : 0.0f; afin[1] = 0.0f;

    float minv[8];
    int   mini[8];
    #pragma unroll
    for (int r = 0; r < 8; ++r) { minv[r] = 3.4e38f; mini[r] = 0; }

    for (int tc = 0; tc < VQ_NT / CHUNK_T; ++tc) {
        __syncthreads();
        { // cooperative stage of CHUNK_T column tiles of B into LDS
            const float2* src = (const float2*)(bp + tc * CHUNK_FLOATS);
            float2*       dst = (float2*)lds;
            for (int i = tid; i < CHUNK_FLOATS / 2; i += 256) dst[i] = src[i];
        }
        __syncthreads();

        #pragma unroll
        for (int tl = 0; tl < CHUNK_T; ++tl) {
            const int t = tc * CHUNK_T + tl;
            const v2f* bfrag = (const v2f*)(lds + tl * FRAG_FLOATS);
            v8f c = {0.f, 0.f, 0.f, 0.f, 0.f, 0.f, 0.f, 0.f};
            #pragma unroll
            for (int s = 0; s < 16; ++s) {
                v2f b = bfrag[s * 32 + lane];
                c = __builtin_amdgcn_wmma_f32_16x16x4_f32(
                        false, a[s], false, b, (short)0, c, false, false);
            }
            {   // ||e||^2 row
                v2f b = bfrag[16 * 32 + lane];
                c = __builtin_amdgcn_wmma_f32_16x16x4_f32(
                        false, afin, false, b, (short)0, c, false, false);
            }
            // c[r] = dist(row m0 + r + 8*hi, code t*16 + ln)
            const int nidx = t * 16 + ln;
            #pragma unroll
            for (int r = 0; r < 8; ++r) {
                bool better = c[r] < minv[r];
                minv[r] = better ? c[r] : minv[r];
                mini[r] = better ? nidx : mini[r];
            }
        }
    }

    // argmin reduce across the 16 lanes sharing each row (tie -> lower index,
    // matching jnp.argmin first-hit semantics)
    #pragma unroll
    for (int r = 0; r < 8; ++r) {
        float v = minv[r]; int i = mini[r];
        #pragma unroll
        for (int off = 1; off <= 8; off <<= 1) {
            float ov = __shfl_xor(v, off, 32);
            int   oi = __shfl_xor(i, off, 32);
            if (ov < v || (ov == v && oi < i)) { v = ov; i = oi; }
        }
        minv[r] = v; mini[r] = i;
    }

    // gather winning embeddings (coalesced: 16 lanes x float2 = 128B, first half of a 256B row)
    #pragma unroll
    for (int r = 0; r < 8; ++r) {
        const int orow = m0 + r + 8 * hi;
        const int idx  = mini[r];
        const float2 q = *(const float2*)(emb + (size_t)idx * VQ_D + 2 * ln);
        *(float2*)(out + (size_t)orow * VQ_D + 2 * ln) = q;
        if (ln == 0) atomicAdd(&counts[idx], 1u);
    }
}

// ---------------------------------------------------------------------------
// Perplexity: exp(-sum p*log(p+eps)), p = counts/N. One 1024-thread block.
// ---------------------------------------------------------------------------
__global__ void vq_perplexity(const unsigned int* __restrict__ counts,
                              float* __restrict__ out_perp, float invN) {
    __shared__ float wsum[32];
    const int tid = threadIdx.x;
    float p = (float)counts[tid] * invN;
    float v = p * logf(p + 1e-10f);
    #pragma unroll
    for (int off = 16; off > 0; off >>= 1) v += __shfl_xor(v, off, 32);
    if ((tid & 31) == 0) wsum[tid >> 5] = v;
    __syncthreads();
    if (tid < 32) {
        float s = wsum[tid];
        #pragma unroll
        for (int off = 16; off > 0; off >>= 1) s += __shfl_xor(s, off, 32);
        if (tid == 0) *out_perp = expf(-s);
    }
}

extern "C" void kernel_launch(void* const* d_in, const int* in_sizes, int n_in,
                              void* d_out, int out_size, void* d_ws, size_t ws_size,
                              hipStream_t stream) {
    const float* x   = (const float*)d_in[0];   // [N, 64] fp32 (flattened)
    const float* emb = (const float*)d_in[1];   // [1024, 64] fp32
    float* out = (float*)d_out;                 // N*64 quantized + 1 perplexity

    const int N = in_sizes[0] / VQ_D;           // 65536

    float*        bp     = (float*)d_ws;
    unsigned int* counts = (unsigned int*)((char*)d_ws + (size_t)BP_FLOATS * sizeof(float));

    vq_prep_b<<<(BP_FLOATS + 255) / 256, 256, 0, stream>>>(emb, bp);
    vq_zero_counts<<<(VQ_K + 255) / 256, 256, 0, stream>>>(counts);
    vq_main<<<N / 128, 256, 0, stream>>>(x, emb, bp, out, counts);
    vq_perplexity<<<1, VQ_K, 0, stream>>>(counts, out + (size_t)N * VQ_D,
                                          1.0f / (float)N);
}